// GRUDecoder_58832462021289
// MI455X (gfx1250) — compile-verified
//
#include <hip/hip_runtime.h>
#include <math.h>

// GRU decoder for MI455X (gfx1250, wave32, WMMA + async LDS copies).
// B=16, W=64 -> T=1024 steps; D=256, H=512, VOCAB=28996.

typedef float v2f __attribute__((ext_vector_type(2)));
typedef float v8f __attribute__((ext_vector_type(8)));

#define T_STEPS 1024
#define D_IN    256
#define H_DIM   512
#define G3      1536          // 3*H
#define VOCAB   28996
#define NSCAN   32            // workgroups cooperating on the scan
#define JPW     16            // h elements per scan workgroup (H/NSCAN)
#define RPW     48            // w_hh rows per scan workgroup (3*JPW)

// ---------------------------------------------------------------------------
// Zero the per-step arrival counters (must run every launch; graph replays).
// ---------------------------------------------------------------------------
__global__ void zero_u32(unsigned* __restrict__ p, int n) {
  int i = blockIdx.x * blockDim.x + threadIdx.x;
  if (i < n) p[i] = 0u;
}

// ---------------------------------------------------------------------------
// Async global->LDS copy, 16 bytes per lane (GLOBAL_LOAD_ASYNC_TO_LDS_B128,
// tracked by ASYNCcnt). lds_off = byte offset inside LDS (low 32 bits of the
// flat shared address), g = per-lane global source (16B aligned).
// ---------------------------------------------------------------------------
__device__ __forceinline__ void async_b128(unsigned lds_off, const float* g) {
  asm volatile("global_load_async_to_lds_b128 %0, %1, off"
               : : "v"(lds_off), "v"(g) : "memory");
}

// ---------------------------------------------------------------------------
// C[M,N] = X[M,K] @ W[N,K]^T + bias[N], f32 via V_WMMA_F32_16X16X4_F32.
// Block tile 64(M) x 128(N), K-chunk 32, double-buffered LDS filled by async
// copies (6 per thread per chunk -> s_wait_asynccnt 6 overlaps the prefetch
// of chunk i+1 with the WMMA stream on chunk i). Row stride 36 floats gives
// conflict-free 16x16x4 fragment reads. 256 threads = 8 waves; wave (wm,wn)
// owns a 16x64 strip = four 16x16 WMMA accumulators.
// Requires: M % 64 == 0, K % 32 == 0. N is guarded (tail tiles OK: clamped
// async rows only feed accumulator columns that are never stored).
// ---------------------------------------------------------------------------
#define LSTR 36
__global__ __launch_bounds__(256) void gemm_nt_wmma(
    const float* __restrict__ X, const float* __restrict__ W,
    const float* __restrict__ bias, float* __restrict__ C,
    int M, int N, int K)
{
  __shared__ __align__(16) float Asm[2][64 * LSTR];
  __shared__ __align__(16) float Bsm[2][128 * LSTR];

  const int tid   = threadIdx.x;
  const int lane  = tid & 31;
  const int wave  = tid >> 5;
  const int wm    = wave >> 1;       // 0..3  (16-row strip)
  const int wn    = wave & 1;        // 0..1  (64-col strip)
  const int m0    = blockIdx.y * 64;
  const int n0    = blockIdx.x * 128;
  const int l16   = lane & 15;
  const int lhalf = lane >> 4;       // 0: K even pair, 1: K odd pair

  v8f acc[4] = {};

  // Stage one 32-wide K chunk into buffer `buf` with async LDS copies:
  // A: 64 rows x 8 segs (16B) = 2/thread; B: 128 rows x 8 segs = 4/thread.
  auto stage = [&](int buf, int kc) {
#pragma unroll
    for (int s = tid; s < 64 * 8; s += 256) {
      int r = s >> 3, c4 = (s & 7) * 4;
      async_b128((unsigned)(uintptr_t)&Asm[buf][r * LSTR + c4],
                 &X[(m0 + r) * K + kc + c4]);
    }
#pragma unroll
    for (int s = tid; s < 128 * 8; s += 256) {
      int r = s >> 3, c4 = (s & 7) * 4;
      int n = n0 + r;
      if (n >= N) n = N - 1;   // keep address legal; column never stored
      async_b128((unsigned)(uintptr_t)&Bsm[buf][r * LSTR + c4],
                 &W[n * K + kc + c4]);
    }
  };

  stage(0, 0);

  for (int kc = 0; kc < K; kc += 32) {
    const int buf = (kc >> 5) & 1;
    const bool hasnext = (kc + 32) < K;
    if (hasnext) {
      stage(buf ^ 1, kc + 32);                    // prefetch next chunk
      asm volatile("s_wait_asynccnt 0x6" ::: "memory");  // chunk kc landed
    } else {
      asm volatile("s_wait_asynccnt 0x0" ::: "memory");
    }
    __syncthreads();

#pragma unroll
    for (int k2 = 0; k2 < 32; k2 += 4) {
      // A fragment (16x4): lanes 0-15 hold K=k2,k2+1; lanes 16-31 K=k2+2,k2+3
      v2f a = *(const v2f*)&Asm[buf][(wm * 16 + l16) * LSTR + k2 + 2 * lhalf];
#pragma unroll
      for (int s = 0; s < 4; ++s) {
        v2f b = *(const v2f*)&Bsm[buf][(wn * 64 + s * 16 + l16) * LSTR + k2 + 2 * lhalf];
        acc[s] = __builtin_amdgcn_wmma_f32_16x16x4_f32(
            /*neg_a=*/false, a, /*neg_b=*/false, b,
            /*c_mod=*/(short)0, acc[s], /*reuse_a=*/false, /*reuse_b=*/false);
      }
    }
    __syncthreads();   // all waves done with `buf` before chunk kc+64 lands in it
  }

  // Epilogue: D VGPR v -> row m0+wm*16+v+8*lhalf, col = n (lane 0-15 == N).
  for (int s = 0; s < 4; ++s) {
    int n = n0 + wn * 64 + s * 16 + l16;
    if (n < N) {
      float bv = bias[n];
#pragma unroll
      for (int v = 0; v < 8; ++v) {
        int m = m0 + wm * 16 + v + 8 * lhalf;
        C[(size_t)m * N + n] = acc[s][v] + bv;
      }
    }
  }
}

// ---------------------------------------------------------------------------
// Sequential GRU scan. 32 persistent workgroups; WG g owns h[16g..16g+16) and
// keeps the matching 48 rows of w_hh (96 KB) in LDS. Per step: local 48x512
// matvec against shared h, gate math for its 16 elements, write hs[t], then a
// device-scope split barrier (per-step L2 atomic counter) before the next step.
// ---------------------------------------------------------------------------
__global__ __launch_bounds__(256) void gru_scan(
    const float* __restrict__ xg,      // [T, 3H]  (x @ w_ih^T + b_ih)
    const float* __restrict__ h0,      // [H]
    const float* __restrict__ w_hh,    // [3H, H]
    const float* __restrict__ b_hh,    // [3H]
    float* __restrict__ hs,            // [T, H]
    unsigned* __restrict__ arrive)     // [T], zeroed before launch
{
  extern __shared__ float smem[];
  float* wsl = smem;                   // RPW * H_DIM
  float* bhl = wsl + RPW * H_DIM;      // RPW
  float* hl  = bhl + RPW;              // H_DIM
  float* ghl = hl + H_DIM;             // RPW

  const int tid   = threadIdx.x;
  const int g     = blockIdx.x;        // 0..31
  const int jbase = g * JPW;

  // Preload this WG's 48 rows of w_hh (rows j, H+j, 2H+j for j in slice).
  for (int i = tid; i < RPW * H_DIM; i += 256) {
    int rl = i >> 9, c = i & (H_DIM - 1);
    int grow = (rl >> 4) * H_DIM + jbase + (rl & 15);
    wsl[i] = w_hh[grow * H_DIM + c];
  }
  if (tid < RPW) {
    int grow = (tid >> 4) * H_DIM + jbase + (tid & 15);
    bhl[tid] = b_hh[grow];
  }
  __syncthreads();

  const int row = tid >> 2;            // 0..63 (rows 0..47 active)
  const int q   = tid & 3;             // quarter of the 512-dot

  for (int t = 0; t < T_STEPS; ++t) {
    // Bring current h into LDS (previous step's full vector, or h0).
    const float* hsrc = (t == 0) ? h0 : (hs + (size_t)(t - 1) * H_DIM);
    hl[tid]       = hsrc[tid];
    hl[256 + tid] = hsrc[256 + tid];
    __syncthreads();

    // gh = w_hh_slice @ h : 4 lanes per row, 128-wide partials, xor-reduce.
    if (row < RPW) {
      const float* wr = wsl + row * H_DIM + q * 128;
      const float* hv = hl + q * 128;
      float s = 0.0f;
#pragma unroll 8
      for (int c = 0; c < 128; ++c) s = fmaf(wr[c], hv[c], s);
      s += __shfl_xor(s, 1, 32);
      s += __shfl_xor(s, 2, 32);
      if (q == 0) ghl[row] = s;
    }
    __syncthreads();

    // Gates for this WG's 16 h elements.
    if (tid < JPW) {
      int j = jbase + tid;
      const float* xgt = xg + (size_t)t * G3;
      float xr = xgt[j], xz = xgt[H_DIM + j], xn = xgt[2 * H_DIM + j];
      float hr = ghl[tid]            + bhl[tid];
      float hz = ghl[JPW + tid]      + bhl[JPW + tid];
      float hn = ghl[2 * JPW + tid]  + bhl[2 * JPW + tid];
      float r = 1.0f / (1.0f + __expf(-(xr + hr)));
      float z = 1.0f / (1.0f + __expf(-(xz + hz)));
      float n = tanhf(xn + r * hn);
      float hnew = (1.0f - z) * n + z * hl[j];
      hs[(size_t)t * H_DIM + j] = hnew;
    }

    // Device-scope split barrier across the 32 workgroups for step t.
    __threadfence();                    // release: hs[t] slice visible
    __syncthreads();
    if (tid == 0) {
      __hip_atomic_fetch_add(&arrive[t], 1u, __ATOMIC_ACQ_REL,
                             __HIP_MEMORY_SCOPE_AGENT);
      while (__hip_atomic_load(&arrive[t], __ATOMIC_ACQUIRE,
                               __HIP_MEMORY_SCOPE_AGENT) < (unsigned)NSCAN) {
        __builtin_amdgcn_s_sleep(1);
      }
    }
    __syncthreads();
    __threadfence();                    // acquire: see other WGs' hs[t] slices
  }
}

// ---------------------------------------------------------------------------
// Host-side orchestration (graph-capture safe: only kernel launches).
// Workspace layout: xg [6 MB] | hs [2 MB] | arrive [4 KB].
// ---------------------------------------------------------------------------
extern "C" void kernel_launch(void* const* d_in, const int* in_sizes, int n_in,
                              void* d_out, int out_size, void* d_ws, size_t ws_size,
                              hipStream_t stream) {
  (void)in_sizes; (void)n_in; (void)out_size; (void)ws_size;

  const float* x      = (const float*)d_in[0];  // [16,64,256]
  const float* enc    = (const float*)d_in[1];  // [1,512]
  const float* w_ih   = (const float*)d_in[2];  // [1536,256]
  const float* w_hh   = (const float*)d_in[3];  // [1536,512]
  const float* b_ih   = (const float*)d_in[4];  // [1536]
  const float* b_hh   = (const float*)d_in[5];  // [1536]
  const float* w_pred = (const float*)d_in[6];  // [28996,512]
  const float* b_pred = (const float*)d_in[7];  // [28996]
  float* out = (float*)d_out;                   // [1024, 28996]

  char* ws = (char*)d_ws;
  float*    xgbuf  = (float*)ws;                                   // 6291456 B
  float*    hsbuf  = (float*)(ws + (size_t)T_STEPS * G3 * 4);      // 2097152 B
  unsigned* arrive = (unsigned*)(ws + (size_t)T_STEPS * G3 * 4
                                    + (size_t)T_STEPS * H_DIM * 4); // 4096 B

  // 1) zero the scan barrier counters (every launch / graph replay)
  zero_u32<<<4, 256, 0, stream>>>(arrive, T_STEPS);

  // 2) xg = x @ w_ih^T + b_ih   (M=1024, N=1536, K=256)
  gemm_nt_wmma<<<dim3(G3 / 128, T_STEPS / 64), 256, 0, stream>>>(
      x, w_ih, b_ih, xgbuf, T_STEPS, G3, D_IN);

  // 3) sequential GRU scan over 1024 steps (32 cooperating WGs)
  size_t scan_smem = (size_t)(RPW * H_DIM + RPW + H_DIM + RPW) * sizeof(float);
  gru_scan<<<NSCAN, 256, scan_smem, stream>>>(xgbuf, enc, w_hh, b_hh, hsbuf, arrive);

  // 4) preds = hs @ w_pred^T + b_pred  (M=1024, N=28996, K=512)
  gemm_nt_wmma<<<dim3((VOCAB + 127) / 128, T_STEPS / 64), 256, 0, stream>>>(
      hsbuf, w_pred, b_pred, out, T_STEPS, VOCAB, H_DIM);
}